// CrissCrossAttention_77197742178884
// MI455X (gfx1250) — compile-verified
//
#include <hip/hip_runtime.h>
#include <math.h>

#define BB 4
#define CC 64
#define CQn 8
#define HH 128
#define WWD 128
#define NN (HH*WWD)   // 16384
#define LP 132        // padded LDS row stride (floats): 132 % 64 = 4 -> conflict-free b64

typedef __attribute__((ext_vector_type(2))) float v2f;
typedef __attribute__((ext_vector_type(8))) float v8f;

__device__ __forceinline__ v8f wmma4(v2f a, v2f b, v8f c) {
  // D = A(16x4) * B(4x16) + C, fp32 WMMA
  return __builtin_amdgcn_wmma_f32_16x16x4_f32(false, a, false, b, (short)0, c, false, false);
}

// CDNA5 async global->LDS copy (ASYNCcnt path, bypasses VGPRs).
// Generic shared pointers carry the LDS byte offset in their low 32 bits
// (flat aperture = high 32 bits only), which is exactly what VDST wants.
__device__ __forceinline__ void async_ld32(float* lds_dst, const float* gsrc) {
  unsigned off = (unsigned)(uintptr_t)lds_dst;
  asm volatile("global_load_async_to_lds_b32 %0, %1, off"
               :: "v"(off), "v"(gsrc) : "memory");
}
__device__ __forceinline__ void async_wait0() {
  asm volatile("s_wait_asynccnt 0x0" ::: "memory");
}

// ---------------------------------------------------------------------------
// 1x1 conv as GEMM: Out[b,o,n] = sum_i W[o,i] * X[b,i,n] + bias[o]
// ABSGELU variant: input = |X - X2|, epilogue = exact GELU.
// ---------------------------------------------------------------------------
template<int OCH, bool ABSGELU>
__global__ __launch_bounds__(128)
void conv1x1_k(const float* __restrict__ X, const float* __restrict__ X2,
               const float* __restrict__ Wm, const float* __restrict__ bias,
               float* __restrict__ Out)
{
  constexpr int OP = (OCH < 16) ? 16 : OCH;
  __shared__ float lw[OP * CC];
  for (int idx = threadIdx.x; idx < OP * CC; idx += blockDim.x) {
    int o = idx / CC;
    lw[idx] = (o < OCH) ? Wm[idx] : 0.f;   // zero-pad rows for O=8
  }
  __syncthreads();

  const int lane = threadIdx.x & 31;
  const int wv   = threadIdx.x >> 5;
  const int ml   = lane & 15;
  const int kh   = (lane >> 4) << 1;       // 0 (lanes 0-15) or 2 (lanes 16-31)

  for (int t = 0; t < 4; ++t) {
    int tile = blockIdx.x * 16 + wv * 4 + t;   // 4096 n-tiles total
    int b    = tile >> 10;                     // 1024 tiles per batch
    int n0   = (tile & 1023) << 4;
    const float* Xb  = X  + (size_t)b * CC * NN;
    const float* X2b = ABSGELU ? (X2 + (size_t)b * CC * NN) : nullptr;
    int n = n0 + ml;

    for (int m0 = 0; m0 < OCH; m0 += 16) {
      v8f acc = {};
      for (int k0 = 0; k0 < CC; k0 += 4) {
        int kk = k0 + kh;
        v2f a; a.x = lw[(m0 + ml) * CC + kk];
               a.y = lw[(m0 + ml) * CC + kk + 1];
        v2f bf;
        if (ABSGELU) {
          bf.x = fabsf(Xb[(size_t)kk * NN + n]       - X2b[(size_t)kk * NN + n]);
          bf.y = fabsf(Xb[(size_t)(kk + 1) * NN + n] - X2b[(size_t)(kk + 1) * NN + n]);
        } else {
          bf.x = Xb[(size_t)kk * NN + n];
          bf.y = Xb[(size_t)(kk + 1) * NN + n];
        }
        acc = wmma4(a, bf, acc);
      }
      int rbase = m0 + ((lane >> 4) << 3);
      #pragma unroll
      for (int r = 0; r < 8; ++r) {
        int o = rbase + r;
        if (o < OCH) {
          float vv = acc[r] + bias[o];
          if (ABSGELU) vv = 0.5f * vv * (1.f + erff(vv * 0.70710678118654752f));
          Out[((size_t)b * OCH + o) * NN + n] = vv;
        }
      }
    }
  }
}

// ---------------------------------------------------------------------------
// Column scores: per (b,w): eH[i,j] = sum_c q[c,i,w]*k[c,j,w], mask i==j.
// q/k column tiles (8x128) async-staged into LDS once per block.
// att layout: att[((b*H + h)*W + w)*256 + j], j<128 column part, j>=128 row part
// ---------------------------------------------------------------------------
__global__ __launch_bounds__(256)
void col_scores_k(const float* __restrict__ q, const float* __restrict__ k,
                  float* __restrict__ att)
{
  __shared__ float lq[CQn * LP];
  __shared__ float lk[CQn * LP];
  int bw = blockIdx.x; int b = bw >> 7, w = bw & 127;
  const float* qc = q + (size_t)b * CQn * NN + w;   // qc[c*NN + i*W]
  const float* kc = k + (size_t)b * CQn * NN + w;
  for (int idx = threadIdx.x; idx < CQn * HH; idx += 256) {
    int c = idx >> 7, x = idx & 127;
    async_ld32(&lq[c * LP + x], qc + (size_t)c * NN + (size_t)x * WWD);
    async_ld32(&lk[c * LP + x], kc + (size_t)c * NN + (size_t)x * WWD);
  }
  async_wait0();
  __syncthreads();

  const int lane = threadIdx.x & 31, wv = threadIdx.x >> 5;
  const int ml = lane & 15, kh = (lane >> 4) << 1;
  const int m0 = wv << 4;                           // query row tile (i)

  for (int j0 = 0; j0 < HH; j0 += 16) {
    v8f acc = {};
    for (int k0 = 0; k0 < CQn; k0 += 4) {
      int c = k0 + kh;
      v2f a; a.x = lq[c * LP + m0 + ml];
             a.y = lq[(c + 1) * LP + m0 + ml];
      v2f bf; bf.x = lk[c * LP + j0 + ml];
              bf.y = lk[(c + 1) * LP + j0 + ml];
      acc = wmma4(a, bf, acc);
    }
    int j  = j0 + ml;
    int ib = m0 + ((lane >> 4) << 3);
    #pragma unroll
    for (int r = 0; r < 8; ++r) {
      int i = ib + r;
      float vv = (i == j) ? -__builtin_inff() : acc[r];
      att[((size_t)(b * HH + i) * WWD + w) * 256 + j] = vv;
    }
  }
}

// ---------------------------------------------------------------------------
// Row scores: per (b,h): eW[i,j] = sum_c q[c,h,i]*k[c,h,j]
// ---------------------------------------------------------------------------
__global__ __launch_bounds__(256)
void row_scores_k(const float* __restrict__ q, const float* __restrict__ k,
                  float* __restrict__ att)
{
  __shared__ float lq[CQn * LP];
  __shared__ float lk[CQn * LP];
  int bh = blockIdx.x; int b = bh >> 7, h = bh & 127;
  const float* qr = q + (size_t)b * CQn * NN + (size_t)h * WWD;  // qr[c*NN + w]
  const float* kr = k + (size_t)b * CQn * NN + (size_t)h * WWD;
  for (int idx = threadIdx.x; idx < CQn * HH; idx += 256) {
    int c = idx >> 7, x = idx & 127;
    async_ld32(&lq[c * LP + x], qr + (size_t)c * NN + x);
    async_ld32(&lk[c * LP + x], kr + (size_t)c * NN + x);
  }
  async_wait0();
  __syncthreads();

  const int lane = threadIdx.x & 31, wv = threadIdx.x >> 5;
  const int ml = lane & 15, kh = (lane >> 4) << 1;
  const int m0 = wv << 4;

  for (int j0 = 0; j0 < WWD; j0 += 16) {
    v8f acc = {};
    for (int k0 = 0; k0 < CQn; k0 += 4) {
      int c = k0 + kh;
      v2f a; a.x = lq[c * LP + m0 + ml];
             a.y = lq[(c + 1) * LP + m0 + ml];
      v2f bf; bf.x = lk[c * LP + j0 + ml];
              bf.y = lk[(c + 1) * LP + j0 + ml];
      acc = wmma4(a, bf, acc);
    }
    int j  = j0 + ml;
    int ib = m0 + ((lane >> 4) << 3);
    #pragma unroll
    for (int r = 0; r < 8; ++r) {
      int i = ib + r;
      att[((size_t)bh * WWD + i) * 256 + 128 + j] = acc[r];
    }
  }
}

// ---------------------------------------------------------------------------
// Joint softmax over 256 logits per pixel; one wave32 per pixel.
// ---------------------------------------------------------------------------
__global__ __launch_bounds__(256)
void softmax_k(float* __restrict__ att)
{
  const int lane = threadIdx.x & 31, wv = threadIdx.x >> 5;
  int pix = blockIdx.x * 8 + wv;                 // BB*HH*WWD pixels
  float* a = att + (size_t)pix * 256;
  float v[8];
  float mx = -3.4e38f;
  #pragma unroll
  for (int t = 0; t < 8; ++t) { v[t] = a[t * 32 + lane]; mx = fmaxf(mx, v[t]); }
  #pragma unroll
  for (int off = 16; off > 0; off >>= 1) mx = fmaxf(mx, __shfl_xor(mx, off, 32));
  float s = 0.f;
  #pragma unroll
  for (int t = 0; t < 8; ++t) { v[t] = __expf(v[t] - mx); s += v[t]; }
  #pragma unroll
  for (int off = 16; off > 0; off >>= 1) s += __shfl_xor(s, off, 32);
  float inv = 1.f / s;
  #pragma unroll
  for (int t = 0; t < 8; ++t) a[t * 32 + lane] = v[t] * inv;
}

// ---------------------------------------------------------------------------
// Column apply: per (b,w): out[c,i] = sum_j v[c,j,w] * attH[i,j]   (writes)
// V column (64x128) async-staged to LDS; 4 channel-tile accumulators per wave
// so each att probability pair is loaded once (b64) and feeds 4 WMMAs.
// ---------------------------------------------------------------------------
__global__ __launch_bounds__(256)
void col_apply_k(const float* __restrict__ v, const float* __restrict__ att,
                 float* __restrict__ out)
{
  __shared__ float lv[CC * LP];
  int bw = blockIdx.x; int b = bw >> 7, w = bw & 127;
  const float* vcol = v + (size_t)b * CC * NN + w;   // vcol[c*NN + j*W]
  for (int idx = threadIdx.x; idx < CC * HH; idx += 256) {
    int c = idx >> 7, j = idx & 127;
    async_ld32(&lv[c * LP + j], vcol + (size_t)c * NN + (size_t)j * WWD);
  }
  async_wait0();
  __syncthreads();

  const int lane = threadIdx.x & 31, wvx = threadIdx.x >> 5;
  const int ml = lane & 15, kh = (lane >> 4) << 1;
  const int i = (wvx << 4) + ml;                     // output row index (h)
  const float2* prow =
      (const float2*)(att + ((size_t)(b * HH + i) * WWD + w) * 256);

  v8f a0 = {}, a1 = {}, a2 = {}, a3 = {};
  for (int k0 = 0; k0 < HH; k0 += 4) {
    int j = k0 + kh;
    float2 p2 = prow[j >> 1];
    v2f bf; bf.x = p2.x; bf.y = p2.y;
    v2f f0, f1, f2, f3;
    f0.x = lv[(ml)      * LP + j]; f0.y = lv[(ml)      * LP + j + 1];
    f1.x = lv[(ml + 16) * LP + j]; f1.y = lv[(ml + 16) * LP + j + 1];
    f2.x = lv[(ml + 32) * LP + j]; f2.y = lv[(ml + 32) * LP + j + 1];
    f3.x = lv[(ml + 48) * LP + j]; f3.y = lv[(ml + 48) * LP + j + 1];
    a0 = wmma4(f0, bf, a0);
    a1 = wmma4(f1, bf, a1);
    a2 = wmma4(f2, bf, a2);
    a3 = wmma4(f3, bf, a3);
  }
  int cb = (lane >> 4) << 3;
  #pragma unroll
  for (int r = 0; r < 8; ++r) {
    out[((size_t)(b * CC + cb + r     ) * HH + i) * WWD + w] = a0[r];
    out[((size_t)(b * CC + cb + r + 16) * HH + i) * WWD + w] = a1[r];
    out[((size_t)(b * CC + cb + r + 32) * HH + i) * WWD + w] = a2[r];
    out[((size_t)(b * CC + cb + r + 48) * HH + i) * WWD + w] = a3[r];
  }
}

// ---------------------------------------------------------------------------
// Row apply: per (b,h): out[c,i] += sum_j v[c,h,j] * attW[i,j]   (accumulates)
// ---------------------------------------------------------------------------
__global__ __launch_bounds__(256)
void row_apply_k(const float* __restrict__ v, const float* __restrict__ att,
                 float* __restrict__ out)
{
  __shared__ float lv[CC * LP];
  int bh = blockIdx.x; int b = bh >> 7, h = bh & 127;
  const float* vrow = v + (size_t)b * CC * NN + (size_t)h * WWD;  // vrow[c*NN + j]
  for (int idx = threadIdx.x; idx < CC * HH; idx += 256) {
    int c = idx >> 7, j = idx & 127;
    async_ld32(&lv[c * LP + j], vrow + (size_t)c * NN + j);
  }
  async_wait0();
  __syncthreads();

  const int lane = threadIdx.x & 31, wvx = threadIdx.x >> 5;
  const int ml = lane & 15, kh = (lane >> 4) << 1;
  const int i = (wvx << 4) + ml;                     // output column index (w)
  const float2* prow =
      (const float2*)(att + ((size_t)bh * WWD + i) * 256 + 128);

  v8f a0 = {}, a1 = {}, a2 = {}, a3 = {};
  for (int k0 = 0; k0 < WWD; k0 += 4) {
    int j = k0 + kh;
    float2 p2 = prow[j >> 1];
    v2f bf; bf.x = p2.x; bf.y = p2.y;
    v2f f0, f1, f2, f3;
    f0.x = lv[(ml)      * LP + j]; f0.y = lv[(ml)      * LP + j + 1];
    f1.x = lv[(ml + 16) * LP + j]; f1.y = lv[(ml + 16) * LP + j + 1];
    f2.x = lv[(ml + 32) * LP + j]; f2.y = lv[(ml + 32) * LP + j + 1];
    f3.x = lv[(ml + 48) * LP + j]; f3.y = lv[(ml + 48) * LP + j + 1];
    a0 = wmma4(f0, bf, a0);
    a1 = wmma4(f1, bf, a1);
    a2 = wmma4(f2, bf, a2);
    a3 = wmma4(f3, bf, a3);
  }
  int cb = (lane >> 4) << 3;
  #pragma unroll
  for (int r = 0; r < 8; ++r) {
    out[((size_t)(b * CC + cb + r     ) * HH + h) * WWD + i] += a0[r];
    out[((size_t)(b * CC + cb + r + 16) * HH + h) * WWD + i] += a1[r];
    out[((size_t)(b * CC + cb + r + 32) * HH + h) * WWD + i] += a2[r];
    out[((size_t)(b * CC + cb + r + 48) * HH + h) * WWD + i] += a3[r];
  }
}

// ---------------------------------------------------------------------------
// BatchNorm batch statistics (sum, sumsq per channel), then final fuse.
// ---------------------------------------------------------------------------
__global__ void zero_k(float* s) { if (threadIdx.x < 128) s[threadIdx.x] = 0.f; }

__global__ __launch_bounds__(256)
void stats_k(const float* __restrict__ p, float* __restrict__ st)
{
  int c = blockIdx.x >> 3, chunk = blockIdx.x & 7;
  float s = 0.f, s2 = 0.f;
  for (int t = threadIdx.x; t < 8192; t += 256) {
    int e = chunk * 8192 + t;           // element within the 65536 per channel
    int b = e >> 14, n = e & 16383;
    float vv = p[((size_t)b * CC + c) * NN + n];
    s += vv; s2 += vv * vv;
  }
  __shared__ float r1[256], r2[256];
  r1[threadIdx.x] = s; r2[threadIdx.x] = s2;
  __syncthreads();
  for (int off = 128; off > 0; off >>= 1) {
    if (threadIdx.x < off) {
      r1[threadIdx.x] += r1[threadIdx.x + off];
      r2[threadIdx.x] += r2[threadIdx.x + off];
    }
    __syncthreads();
  }
  if (threadIdx.x == 0) { atomicAdd(&st[c], r1[0]); atomicAdd(&st[64 + c], r2[0]); }
}

__global__ __launch_bounds__(256)
void final_k(const float* __restrict__ p, const float* __restrict__ oA,
             const float* __restrict__ oB, const float* __restrict__ x1,
             const float* __restrict__ x2, const float* __restrict__ st,
             const float* __restrict__ g, const float* __restrict__ bbeta,
             const float* __restrict__ gA, const float* __restrict__ gB,
             float* __restrict__ out)
{
  size_t idx = (size_t)blockIdx.x * 256 + threadIdx.x;   // over B*C*N
  int c = (int)((idx >> 14) & 63);
  const float cnt = (float)(BB * NN);
  float mean = st[c] / cnt;
  float var  = st[64 + c] / cnt - mean * mean;
  float nrm  = (p[idx] - mean) * rsqrtf(var + 1e-5f) * g[c] + bbeta[c];
  out[idx] = gA[0] * oA[idx] + x1[idx] + nrm;
  out[(size_t)BB * CC * NN + idx] = gB[0] * oB[idx] + x2[idx] + nrm;
}

// ---------------------------------------------------------------------------
extern "C" void kernel_launch(void* const* d_in, const int* in_sizes, int n_in,
                              void* d_out, int out_size, void* d_ws, size_t ws_size,
                              hipStream_t stream)
{
  const float* x1  = (const float*)d_in[0];
  const float* x2  = (const float*)d_in[1];
  const float* wqA = (const float*)d_in[2];  const float* bqA = (const float*)d_in[3];
  const float* wkA = (const float*)d_in[4];  const float* bkA = (const float*)d_in[5];
  const float* wvA = (const float*)d_in[6];  const float* bvA = (const float*)d_in[7];
  const float* wqB = (const float*)d_in[8];  const float* bqB = (const float*)d_in[9];
  const float* wkB = (const float*)d_in[10]; const float* bkB = (const float*)d_in[11];
  const float* wvB = (const float*)d_in[12]; const float* bvB = (const float*)d_in[13];
  const float* gA  = (const float*)d_in[14]; const float* gB  = (const float*)d_in[15];
  const float* wp  = (const float*)d_in[16]; const float* bp  = (const float*)d_in[17];
  const float* bng = (const float*)d_in[18]; const float* bnb = (const float*)d_in[19];

  float* ws = (float*)d_ws;
  const size_t SQ = (size_t)BB * CQn * NN;   //   524288 floats
  const size_t SV = (size_t)BB * CC  * NN;   //  4194304 floats
  float* qA  = ws;           float* kA  = qA + SQ;
  float* qB  = kA + SQ;      float* kB  = qB + SQ;
  float* vA  = kB + SQ;      float* vB  = vA + SV;
  float* oA  = vB + SV;      float* oB  = oA + SV;
  float* pb  = oB + SV;      float* st  = pb + SV;
  float* att = st + 128;                      // B*H*W*256 floats

  // 1x1 convs (WMMA GEMMs)
  conv1x1_k<CQn, false><<<256, 128, 0, stream>>>(x2, nullptr, wqA, bqA, qA);
  conv1x1_k<CQn, false><<<256, 128, 0, stream>>>(x1, nullptr, wkA, bkA, kA);
  conv1x1_k<CQn, false><<<256, 128, 0, stream>>>(x1, nullptr, wqB, bqB, qB);
  conv1x1_k<CQn, false><<<256, 128, 0, stream>>>(x1, nullptr, wkB, bkB, kB);
  conv1x1_k<CC,  false><<<256, 128, 0, stream>>>(x1, nullptr, wvA, bvA, vA);
  conv1x1_k<CC,  false><<<256, 128, 0, stream>>>(x1, nullptr, wvB, bvB, vB);

  // Branch A
  col_scores_k<<<BB * WWD, 256, 0, stream>>>(qA, kA, att);
  row_scores_k<<<BB * HH,  256, 0, stream>>>(qA, kA, att);
  softmax_k<<<BB * HH * WWD / 8, 256, 0, stream>>>(att);
  col_apply_k<<<BB * WWD, 256, 0, stream>>>(vA, att, oA);
  row_apply_k<<<BB * HH,  256, 0, stream>>>(vA, att, oA);

  // Branch B (reuses att scratch — stream-ordered)
  col_scores_k<<<BB * WWD, 256, 0, stream>>>(qB, kB, att);
  row_scores_k<<<BB * HH,  256, 0, stream>>>(qB, kB, att);
  softmax_k<<<BB * HH * WWD / 8, 256, 0, stream>>>(att);
  col_apply_k<<<BB * WWD, 256, 0, stream>>>(vB, att, oB);
  row_apply_k<<<BB * HH,  256, 0, stream>>>(vB, att, oB);

  // p = GELU(Wp |oA - oB| + bp), then BN stats and fused finale
  conv1x1_k<CC, true><<<256, 128, 0, stream>>>(oA, oB, wp, bp, pb);
  zero_k<<<1, 128, 0, stream>>>(st);
  stats_k<<<CC * 8, 256, 0, stream>>>(pb, st);
  final_k<<<(int)(SV / 256), 256, 0, stream>>>(pb, oA, oB, x1, x2, st, bng, bnb,
                                               gA, gB, (float*)d_out);
}